// GNNConv_56014963475024
// MI455X (gfx1250) — compile-verified
//
#include <hip/hip_runtime.h>
#include <hip/hip_bf16.h>

#define Nn 100000
#define Dd 128
#define Rr 8
#define Ee 1600000

typedef __attribute__((ext_vector_type(2))) float v2f;
typedef __attribute__((ext_vector_type(8))) float v8f;

// ---------------------------------------------------------------------------
// Kernel 1: zero the workspace (sums [N*R*D] f32, counts [N*R] f32).
// Must run every launch: the scatter kernel accumulates into it.
// ---------------------------------------------------------------------------
__global__ __launch_bounds__(256) void rgcn_zero_ws(float4* __restrict__ p, long n4) {
    long i = (long)blockIdx.x * blockDim.x + threadIdx.x;
    long stride = (long)gridDim.x * blockDim.x;
    float4 z; z.x = 0.f; z.y = 0.f; z.z = 0.f; z.w = 0.f;
    for (; i < n4; i += stride) p[i] = z;
}

// ---------------------------------------------------------------------------
// Kernel 2: edge scatter. One wave (32 lanes) per edge: lane l handles the
// float4 at columns 4l..4l+3 of x[src], atomically accumulated into
// sums[(dst*R+type)*D + ...]; lane 0 bumps the (dst,rel) edge count.
// Coalesced 512B gather per edge; atomics resolve in L2.
// ---------------------------------------------------------------------------
__global__ __launch_bounds__(256) void rgcn_scatter(
    const float* __restrict__ x,
    const int* __restrict__ src,
    const int* __restrict__ dst,
    const int* __restrict__ etype,
    float* __restrict__ sums,
    float* __restrict__ cnts) {
    int lane  = threadIdx.x & 31;
    long wave = ((long)blockIdx.x * blockDim.x + threadIdx.x) >> 5;
    long nw   = ((long)gridDim.x * blockDim.x) >> 5;
    for (long e = wave; e < Ee; e += nw) {
        int s = src[e];
        int d = dst[e];
        int t = etype[e];
        long seg = (long)d * Rr + t;
        const float4 xv = *(const float4*)(x + (long)s * Dd + lane * 4);
        float* sp = sums + seg * Dd + lane * 4;
        atomicAdd(sp + 0, xv.x);
        atomicAdd(sp + 1, xv.y);
        atomicAdd(sp + 2, xv.z);
        atomicAdd(sp + 3, xv.w);
        if (lane == 0) atomicAdd(cnts + seg, 1.0f);
    }
}

// ---------------------------------------------------------------------------
// Kernel 3: fused GEMM. One wave computes a full 16(node) x 128(col) output
// slab = 8 column tiles, so the A fragment (mean, scaled by 1/cnt) is loaded
// and scaled ONCE per k-step and reused by 8 V_WMMA_F32_16X16X4_F32 ops.
// This makes the 410MB `sums` array single-pass from HBM; the 0.6MB of
// weights are re-read from L2/WGP$.
// A(16x4)/B(4x16) fp32 WMMA layout: lane L, VGPR v -> K = k0 + v + 2*(L>>4);
// A row M = L&15, B col N = L&15. C/D: col = L&15, row = v + 8*(L>>4).
// ---------------------------------------------------------------------------
__global__ __launch_bounds__(32) void rgcn_gemm(
    const float* __restrict__ x,
    const float* __restrict__ W,      // [R, D, D] row-major (k rows)
    const float* __restrict__ Wroot,  // [D, D]
    const float* __restrict__ bias,   // [D]
    const float* __restrict__ sums,   // [N*R, D]
    const float* __restrict__ cnts,   // [N*R]
    float* __restrict__ out) {
    const int tm   = blockIdx.x;            // node tile (16 nodes), 0..6249
    const int lane = threadIdx.x;           // 0..31
    const int m    = lane & 15;             // A row / B col / C col
    const int hi   = lane >> 4;             // K half-select
    const int node = tm * 16 + m;

    // 8 accumulators: acc[t] is the 16x16 tile covering cols [16t, 16t+16)
    v8f acc[8];
#pragma unroll
    for (int t = 0; t < 8; ++t) {
        const float bv = bias[16 * t + m];
#pragma unroll
        for (int v = 0; v < 8; ++v) acc[t][v] = bv;
    }

    // --- per-relation mean @ W_r ---
#pragma unroll 1
    for (int r = 0; r < Rr; ++r) {
        const float c   = cnts[(long)node * Rr + r];
        const float inv = 1.0f / fmaxf(c, 1.0f);
        const float* __restrict__ arow = sums + ((long)node * Rr + r) * Dd;
        const float* __restrict__ wmat = W + (long)r * Dd * Dd;
#pragma unroll 4
        for (int k = 0; k < Dd; k += 4) {
            const int ka = k + 2 * hi;
            const float2 av = *(const float2*)(arow + ka);
            v2f a; a.x = av.x * inv; a.y = av.y * inv;
            const float* __restrict__ w0 = wmat + (long)ka * Dd + m;
#pragma unroll
            for (int t = 0; t < 8; ++t) {
                v2f b;
                b.x = w0[16 * t];
                b.y = w0[16 * t + Dd];
                acc[t] = __builtin_amdgcn_wmma_f32_16x16x4_f32(
                    false, a, false, b, (short)0, acc[t], false, false);
            }
        }
    }

    // --- root term: x @ W_root ---
    {
        const float* __restrict__ xrow = x + (long)node * Dd;
#pragma unroll 4
        for (int k = 0; k < Dd; k += 4) {
            const int ka = k + 2 * hi;
            const float2 av = *(const float2*)(xrow + ka);
            v2f a; a.x = av.x; a.y = av.y;
            const float* __restrict__ w0 = Wroot + (long)ka * Dd + m;
#pragma unroll
            for (int t = 0; t < 8; ++t) {
                v2f b;
                b.x = w0[16 * t];
                b.y = w0[16 * t + Dd];
                acc[t] = __builtin_amdgcn_wmma_f32_16x16x4_f32(
                    false, a, false, b, (short)0, acc[t], false, false);
            }
        }
    }

    // --- store the 16x128 slab: row = v + 8*hi, col = 16t + m ---
#pragma unroll
    for (int t = 0; t < 8; ++t) {
#pragma unroll
        for (int v = 0; v < 8; ++v) {
            const int row = v + 8 * hi;
            out[(long)(tm * 16 + row) * Dd + 16 * t + m] = acc[t][v];
        }
    }
}

// ---------------------------------------------------------------------------
extern "C" void kernel_launch(void* const* d_in, const int* in_sizes, int n_in,
                              void* d_out, int out_size, void* d_ws, size_t ws_size,
                              hipStream_t stream) {
    const float* x     = (const float*)d_in[0];   // [N, D]
    const float* W     = (const float*)d_in[1];   // [R, D, D]
    const float* Wroot = (const float*)d_in[2];   // [D, D]
    const float* bias  = (const float*)d_in[3];   // [D]
    const int*   eidx  = (const int*)d_in[4];     // [2, E]
    const int*   etype = (const int*)d_in[5];     // [E]
    float*       out   = (float*)d_out;           // [N, D]

    const int* src = eidx;        // edge_index[0]
    const int* dst = eidx + Ee;   // edge_index[1]

    float* sums = (float*)d_ws;                   // N*R*D floats
    float* cnts = sums + (long)Nn * Rr * Dd;      // N*R floats

    // 1) zero sums + counts (contiguous in ws)
    const long n4 = ((long)Nn * Rr * Dd + (long)Nn * Rr) / 4;
    rgcn_zero_ws<<<4096, 256, 0, stream>>>((float4*)d_ws, n4);

    // 2) scatter edges into (dst, rel) segment sums + counts
    rgcn_scatter<<<8192, 256, 0, stream>>>(x, src, dst, etype, sums, cnts);

    // 3) fused mean-normalize + 9-way GEMM + bias; one wave per 16x128 slab
    rgcn_gemm<<<Nn / 16, 32, 0, stream>>>(x, W, Wroot, bias, sums, cnts, out);
}